// CDLoss_L1_85478439125594
// MI455X (gfx1250) — compile-verified
//
#include <hip/hip_runtime.h>
#include <math.h>

typedef __attribute__((ext_vector_type(2))) float v2f;
typedef __attribute__((ext_vector_type(8))) float v8f;

// ---------------------------------------------------------------------------
// Build augmented point representations:
//   outQ[i] = (x, y, z, 1)                  -- used as WMMA A-matrix rows
//   outR[i] = (-2x, -2y, -2z, x^2+y^2+z^2)  -- used as WMMA B-matrix columns
// so that A(16x4) * B(4x16) = -2<a,b> + |b|^2 = dist^2 - |a|^2.
// ---------------------------------------------------------------------------
__global__ __launch_bounds__(256) void cd_augment(const float* __restrict__ p,
                                                  float* __restrict__ outQ,
                                                  float* __restrict__ outR,
                                                  int total) {
  int i = blockIdx.x * blockDim.x + threadIdx.x;
  if (i >= total) return;
  float x = p[3 * i + 0], y = p[3 * i + 1], z = p[3 * i + 2];
  float sq = x * x + y * y + z * z;
  outQ[4 * i + 0] = x;         outQ[4 * i + 1] = y;
  outQ[4 * i + 2] = z;         outQ[4 * i + 3] = 1.0f;
  outR[4 * i + 0] = -2.0f * x; outR[4 * i + 1] = -2.0f * y;
  outR[4 * i + 2] = -2.0f * z; outR[4 * i + 3] = sq;
}

// ---------------------------------------------------------------------------
// One wave owns 2 query tiles (32 query points) of one batch and streams all
// reference tiles through V_WMMA_F32_16X16X4_F32, keeping a running column-min
// in the 8 accumulator VGPRs. Per 16x16 tile the loop body is exactly
// 1 WMMA + 8 v_min ops. |a|^2, the zero-clamp and the sqrt are deferred to
// after the full min-reduction (legal: they are row-constant / monotone).
//
// VGPR packing (per ISA 16x4 f32 A layout, B assumed symmetric):
//   lane<16 : {K0,K1} of point (base+lane)
//   lane>=16: {K2,K3} of point (base+lane-16)
// i.e. an 8-byte load at  point*16 + (lane>>4)*8.
// C/D layout: VGPR j -> row (j + 8*(lane>>4)), col (lane&15).
// ---------------------------------------------------------------------------
__global__ __launch_bounds__(256) void cd_nn_pass(
    const float* __restrict__ augQ,   // [B*Nq*4] (x,y,z,1)
    const float* __restrict__ augR,   // [B*Nr*4] (-2x,-2y,-2z,|p|^2)
    const float* __restrict__ rawQ,   // [B*Nq*3]
    float* __restrict__ dOut,         // [B*Nq]   sqrt(min dist^2)
    int B, int Nq, int Nr)
{
  const int wavesPerBlock = blockDim.x >> 5;
  const int wave = blockIdx.x * wavesPerBlock + (threadIdx.x >> 5);
  const int lane = threadIdx.x & 31;
  const int pairTiles = Nq >> 5;            // 32 query points per wave
  const int totalWaves = B * pairTiles;
  if (wave >= totalWaves) return;           // whole-wave exit: EXEC stays all-1s
  const int b  = wave / pairTiles;
  const int qp = wave % pairTiles;
  const int qBase = qp << 5;

  const int pt = lane & 15;
  const int hi = lane >> 4;
  const size_t laneOff = (size_t)pt * 4 + (size_t)hi * 2;

  const float* Aq = augQ + ((size_t)b * Nq + qBase) * 4;
  const float* Rr = augR + (size_t)b * Nr * 4;

  v2f a0 = *(const v2f*)(Aq + laneOff);            // query tile 0
  v2f a1 = *(const v2f*)(Aq + 16 * 4 + laneOff);   // query tile 1

  v8f m0, m1;
#pragma unroll
  for (int j = 0; j < 8; ++j) { m0[j] = 3.402823466e38f; m1[j] = 3.402823466e38f; }

  const int nTilesR = Nr >> 4;
  for (int t = 0; t < nTilesR; t += 2) {
    v2f b0 = *(const v2f*)(Rr + (size_t)t * 64 + laneOff);
    v2f b1 = *(const v2f*)(Rr + (size_t)(t + 1) * 64 + laneOff);
    v8f c00 = {}, c01 = {}, c10 = {}, c11 = {};
    c00 = __builtin_amdgcn_wmma_f32_16x16x4_f32(false, a0, false, b0, (short)0, c00, false, false);
    c10 = __builtin_amdgcn_wmma_f32_16x16x4_f32(false, a1, false, b0, (short)0, c10, false, false);
    c01 = __builtin_amdgcn_wmma_f32_16x16x4_f32(false, a0, false, b1, (short)0, c01, false, false);
    c11 = __builtin_amdgcn_wmma_f32_16x16x4_f32(false, a1, false, b1, (short)0, c11, false, false);
#pragma unroll
    for (int j = 0; j < 8; ++j) {
      m0[j] = fminf(fminf(m0[j], c00[j]), c01[j]);  // fusable to v_min3_num_f32
      m1[j] = fminf(fminf(m1[j], c10[j]), c11[j]);
    }
  }

  // min across the 16 lanes holding each row's columns (wave32, xor<=8 stays
  // inside each 16-lane half).
#pragma unroll
  for (int j = 0; j < 8; ++j) {
    float v0 = m0[j], v1 = m1[j];
#pragma unroll
    for (int s = 1; s <= 8; s <<= 1) {
      v0 = fminf(v0, __shfl_xor(v0, s, 32));
      v1 = fminf(v1, __shfl_xor(v1, s, 32));
    }
    m0[j] = v0; m1[j] = v1;
  }

  if (pt == 0) {  // lanes 0 and 16 finalize rows j+0..7 / j+8..15 of each tile
    const float* q = rawQ + ((size_t)b * Nq + qBase) * 3;
    float* o = dOut + (size_t)b * Nq + qBase;
#pragma unroll
    for (int j = 0; j < 8; ++j) {
      int r0 = j + hi * 8;
      float x = q[r0 * 3], y = q[r0 * 3 + 1], z = q[r0 * 3 + 2];
      o[r0] = sqrtf(fmaxf(m0[j] + x * x + y * y + z * z, 0.0f));
      int r1 = 16 + j + hi * 8;
      x = q[r1 * 3]; y = q[r1 * 3 + 1]; z = q[r1 * 3 + 2];
      o[r1] = sqrtf(fmaxf(m1[j] + x * x + y * y + z * z, 0.0f));
    }
  }
}

// ---------------------------------------------------------------------------
// Final scalar: 0.5 * (mean(d1) + mean(d2)). Single block, LDS tree reduce.
// ---------------------------------------------------------------------------
__global__ __launch_bounds__(256) void cd_finalize(const float* __restrict__ d1,
                                                   const float* __restrict__ d2,
                                                   int n1, int n2,
                                                   float* __restrict__ out) {
  __shared__ float s1[256];
  __shared__ float s2[256];
  float a = 0.0f, c = 0.0f;
  for (int i = threadIdx.x; i < n1; i += blockDim.x) a += d1[i];
  for (int i = threadIdx.x; i < n2; i += blockDim.x) c += d2[i];
  s1[threadIdx.x] = a; s2[threadIdx.x] = c;
  __syncthreads();
  for (int off = blockDim.x >> 1; off > 0; off >>= 1) {
    if ((int)threadIdx.x < off) {
      s1[threadIdx.x] += s1[threadIdx.x + off];
      s2[threadIdx.x] += s2[threadIdx.x + off];
    }
    __syncthreads();
  }
  if (threadIdx.x == 0)
    out[0] = 0.5f * (s1[0] / (float)n1 + s2[0] / (float)n2);
}

extern "C" void kernel_launch(void* const* d_in, const int* in_sizes, int n_in,
                              void* d_out, int out_size, void* d_ws, size_t ws_size,
                              hipStream_t stream) {
  const float* p1 = (const float*)d_in[0];
  const float* p2 = (const float*)d_in[1];
  float* out = (float*)d_out;

  const int B = 8;
  const int N = in_sizes[0] / (B * 3);
  const int M = in_sizes[1] / (B * 3);

  // Workspace layout (floats): ~4.7 MB total
  float* ws    = (float*)d_ws;
  float* aug1q = ws;                               // B*N*4  (x,y,z,1) of p1
  float* aug1r = aug1q + (size_t)B * N * 4;        // B*N*4  (-2x..,|p1|^2)
  float* aug2q = aug1r + (size_t)B * N * 4;        // B*M*4  (x,y,z,1) of p2
  float* aug2r = aug2q + (size_t)B * M * 4;        // B*M*4  (-2x..,|p2|^2)
  float* d1    = aug2r + (size_t)B * M * 4;        // B*N
  float* d2    = d1 + (size_t)B * N;               // B*M

  const int t1 = B * N, t2 = B * M;
  cd_augment<<<(t1 + 255) / 256, 256, 0, stream>>>(p1, aug1q, aug1r, t1);
  cd_augment<<<(t2 + 255) / 256, 256, 0, stream>>>(p2, aug2q, aug2r, t2);

  // Direction 1: queries = p1, references = p2
  {
    int waves = B * (N >> 5);
    int blocks = (waves * 32 + 255) / 256;
    cd_nn_pass<<<blocks, 256, 0, stream>>>(aug1q, aug2r, p1, d1, B, N, M);
  }
  // Direction 2: queries = p2, references = p1
  {
    int waves = B * (M >> 5);
    int blocks = (waves * 32 + 255) / 256;
    cd_nn_pass<<<blocks, 256, 0, stream>>>(aug2q, aug1r, p2, d2, B, M, N);
  }

  cd_finalize<<<1, 256, 0, stream>>>(d1, d2, t1, t2, out);
}